// NaMixedOp_446676599404
// MI455X (gfx1250) — compile-verified
//
#include <hip/hip_runtime.h>
#include <math.h>

#define NN 50000
#define NE 800000
#define DIM 64

typedef __attribute__((ext_vector_type(2))) float v2f;
typedef __attribute__((ext_vector_type(8))) float v8f;
typedef long long ll;

// ---------------- small helpers ----------------

__device__ __forceinline__ void atomicMaxF(float* addr, float val) {
    unsigned int* ua = (unsigned int*)addr;
    unsigned int old = __float_as_uint(*addr);
    while (__uint_as_float(old) < val) {
        unsigned int assumed = old;
        old = atomicCAS(ua, assumed, __float_as_uint(val));
        if (old == assumed) break;
    }
}

__global__ void k_fill(float* p, float v, int n) {
    int i = blockIdx.x * blockDim.x + threadIdx.x;
    if (i < n) p[i] = v;
}

// ---------------- WMMA fp32 GEMM: C[n x 64] (+)= A[n x 64] @ W[64 x 64] (+ bias) ----------------
// one wave per 16-row stripe; 4 column tiles of 16; K stepped by 4 (V_WMMA_F32_16X16X4_F32)

__global__ __launch_bounds__(256) void k_gemm64(const float* __restrict__ A,
                                                const float* __restrict__ W,
                                                const float* __restrict__ bias,
                                                float* __restrict__ C,
                                                int n, int accum) {
    __shared__ float sW[64 * 66];  // transposed: sW[col*66 + k], pad 66 -> 8B-aligned pairs, no bank conflict
    for (int i = threadIdx.x; i < 64 * 64; i += 256) {
        int k = i >> 6, c = i & 63;
        sW[c * 66 + k] = W[i];
    }
    __syncthreads();

    const int wave = threadIdx.x >> 5;
    const int lane = threadIdx.x & 31;
    const int row0 = (blockIdx.x * 8 + wave) * 16;
    if (row0 >= n) return;  // uniform per wave: EXEC stays all-1s for WMMA

    const int half = lane >> 4;   // 0: K pair {0,1}; 1: K pair {2,3}
    const int l15  = lane & 15;
    const int kb   = half * 2;

    v8f acc0 = {}, acc1 = {}, acc2 = {}, acc3 = {};
    const float* arow = A + (size_t)(row0 + l15) * DIM;

#pragma unroll
    for (int k0 = 0; k0 < DIM; k0 += 4) {
        v2f a  = *(const v2f*)(arow + k0 + kb);
        v2f b0 = *(const v2f*)(&sW[( 0 + l15) * 66 + k0 + kb]);
        v2f b1 = *(const v2f*)(&sW[(16 + l15) * 66 + k0 + kb]);
        v2f b2 = *(const v2f*)(&sW[(32 + l15) * 66 + k0 + kb]);
        v2f b3 = *(const v2f*)(&sW[(48 + l15) * 66 + k0 + kb]);
        acc0 = __builtin_amdgcn_wmma_f32_16x16x4_f32(false, a, false, b0, (short)0, acc0, false, false);
        acc1 = __builtin_amdgcn_wmma_f32_16x16x4_f32(false, a, false, b1, (short)0, acc1, false, false);
        acc2 = __builtin_amdgcn_wmma_f32_16x16x4_f32(false, a, false, b2, (short)0, acc2, false, false);
        acc3 = __builtin_amdgcn_wmma_f32_16x16x4_f32(false, a, false, b3, (short)0, acc3, false, false);
    }

    const bool hasb = (bias != nullptr);
    const int rbase = row0 + half * 8;
#pragma unroll
    for (int v = 0; v < 8; ++v) {
        float* crow = C + (size_t)(rbase + v) * DIM;
        float vals[4] = {acc0[v], acc1[v], acc2[v], acc3[v]};
#pragma unroll
        for (int t = 0; t < 4; ++t) {
            int c = t * 16 + l15;
            float val = vals[t] + (hasb ? bias[c] : 0.0f);
            if (accum) crow[c] += val; else crow[c] = val;
        }
    }
}

// ---------------- edge kernels ----------------

__global__ void k_deg_nsum(const ll* __restrict__ src, const ll* __restrict__ dst,
                           const float* __restrict__ x, float* deg, float* nsum, int E) {
    int e = blockIdx.x * blockDim.x + threadIdx.x;
    if (e >= E) return;
    int s = (int)src[e], d = (int)dst[e];
    atomicAdd(&deg[d], 1.0f);
    const float4* xs = (const float4*)(x + (size_t)s * DIM);
    float* nd = nsum + (size_t)d * DIM;
#pragma unroll
    for (int i = 0; i < 16; ++i) {
        float4 v = xs[i];
        atomicAdd(&nd[4 * i + 0], v.x);
        atomicAdd(&nd[4 * i + 1], v.y);
        atomicAdd(&nd[4 * i + 2], v.z);
        atomicAdd(&nd[4 * i + 3], v.w);
    }
}

// out[d,:] += wn[s]*wn[d] * t[s,:]
__global__ void k_scatter_norm(const ll* __restrict__ src, const ll* __restrict__ dst,
                               const float* __restrict__ t, const float* __restrict__ wn,
                               float* out, int E) {
    int e = blockIdx.x * blockDim.x + threadIdx.x;
    if (e >= E) return;
    int s = (int)src[e], d = (int)dst[e];
    float c = wn[s] * wn[d];
    const float4* ts = (const float4*)(t + (size_t)s * DIM);
    float* od = out + (size_t)d * DIM;
#pragma unroll
    for (int i = 0; i < 16; ++i) {
        float4 v = ts[i];
        atomicAdd(&od[4 * i + 0], c * v.x);
        atomicAdd(&od[4 * i + 1], c * v.y);
        atomicAdd(&od[4 * i + 2], c * v.z);
        atomicAdd(&od[4 * i + 3], c * v.w);
    }
}

__device__ __forceinline__ void edge_sd(const ll* src, const ll* dst, int e, int E, int& s, int& d) {
    if (e < E) { s = (int)src[e]; d = (int)dst[e]; }
    else       { s = e - E; d = s; }  // appended self loops
}

__device__ __forceinline__ float leaky02(float v) { return v > 0.0f ? v : 0.2f * v; }

__global__ void k_gat_max(const ll* src, const ll* dst, const float* el, const float* er,
                          float* m, int E, int E2) {
    int e = blockIdx.x * blockDim.x + threadIdx.x;
    if (e >= E2) return;
    int s, d; edge_sd(src, dst, e, E, s, d);
    atomicMaxF(&m[d], leaky02(el[s] + er[d]));
}

__global__ void k_gat_sum(const ll* src, const ll* dst, const float* el, const float* er,
                          const float* m, float* z, int E, int E2) {
    int e = blockIdx.x * blockDim.x + threadIdx.x;
    if (e >= E2) return;
    int s, d; edge_sd(src, dst, e, E, s, d);
    atomicAdd(&z[d], expf(leaky02(el[s] + er[d]) - m[d]));
}

__global__ void k_gat_acc(const ll* src, const ll* dst, const float* el, const float* er,
                          const float* m, const float* z, const float* __restrict__ h,
                          float* out, int E, int E2) {
    int e = blockIdx.x * blockDim.x + threadIdx.x;
    if (e >= E2) return;
    int s, d; edge_sd(src, dst, e, E, s, d);
    float alpha = expf(leaky02(el[s] + er[d]) - m[d]) / (z[d] + 1e-16f);
    const float4* hs = (const float4*)(h + (size_t)s * DIM);
    float* od = out + (size_t)d * DIM;
#pragma unroll
    for (int i = 0; i < 16; ++i) {
        float4 v = hs[i];
        atomicAdd(&od[4 * i + 0], alpha * v.x);
        atomicAdd(&od[4 * i + 1], alpha * v.y);
        atomicAdd(&od[4 * i + 2], alpha * v.z);
        atomicAdd(&od[4 * i + 3], alpha * v.w);
    }
}

// ---------------- node / elementwise kernels ----------------

__global__ void k_dinv(const float* deg, float* dinv, float* dinvc, int n) {
    int i = blockIdx.x * blockDim.x + threadIdx.x;
    if (i >= n) return;
    float dg = deg[i];
    dinv[i]  = rsqrtf(dg + 1.0f);
    dinvc[i] = rsqrtf(fmaxf(dg, 1.0f));
}

__global__ void k_nmean(const float* nsum, const float* deg, float* out, int n64) {
    int i = blockIdx.x * blockDim.x + threadIdx.x;
    if (i >= n64) return;
    out[i] = nsum[i] / fmaxf(deg[i >> 6], 1.0f);
}

__global__ void k_addxn(const float* x, const float* nsum, float* out, int n64) {
    int i = blockIdx.x * blockDim.x + threadIdx.x;
    if (i < n64) out[i] = x[i] + nsum[i];
}

__global__ void k_gcn_self(float* o, const float* xw, const float* dinv, const float* b, int n64) {
    int i = blockIdx.x * blockDim.x + threadIdx.x;
    if (i >= n64) return;
    float dv = dinv[i >> 6];
    o[i] += xw[i] * dv * dv + b[i & 63];
}

__global__ void k_negate(float* p, int n64) {
    int i = blockIdx.x * blockDim.x + threadIdx.x;
    if (i < n64) p[i] = -p[i];
}

__global__ void k_t2(float* t2, const float* x, int n64) {  // t2 = -2*t2 - x
    int i = blockIdx.x * blockDim.x + threadIdx.x;
    if (i < n64) t2[i] = -2.0f * t2[i] - x[i];
}

__global__ void k_addbias(float* o, const float* b, int n64) {
    int i = blockIdx.x * blockDim.x + threadIdx.x;
    if (i < n64) o[i] += b[i & 63];
}

__global__ void k_el_er(const float* __restrict__ h, const float* __restrict__ as,
                        const float* __restrict__ ad, float* el, float* er, int n) {
    int i = blockIdx.x * blockDim.x + threadIdx.x;
    if (i >= n) return;
    const float* hr = h + (size_t)i * DIM;
    float s1 = 0.0f, s2 = 0.0f;
#pragma unroll
    for (int j = 0; j < DIM; ++j) { float v = hr[j]; s1 += v * as[j]; s2 += v * ad[j]; }
    el[i] = s1; er[i] = s2;
}

// out (+)= w[k] * elu(o)
__global__ void k_combine(float* out, const float* o, const float* wts, int widx, int first, int n64) {
    int i = blockIdx.x * blockDim.x + threadIdx.x;
    if (i >= n64) return;
    float v = o[i];
    float e = v > 0.0f ? v : (expf(v) - 1.0f);
    float r = wts[widx] * e;
    if (first) out[i] = r; else out[i] += r;
}

// ---------------- launcher ----------------

extern "C" void kernel_launch(void* const* d_in, const int* in_sizes, int n_in,
                              void* d_out, int out_size, void* d_ws, size_t ws_size,
                              hipStream_t stream) {
    const float* x      = (const float*)d_in[0];
    const float* wts    = (const float*)d_in[1];
    const float* W_mlp  = (const float*)d_in[2];
    const float* b_mlp  = (const float*)d_in[3];
    const float* Wl_sg  = (const float*)d_in[4];
    const float* Wr_sg  = (const float*)d_in[5];
    const float* b_sg   = (const float*)d_in[6];
    const float* W_gcn  = (const float*)d_in[7];
    const float* b_gcn  = (const float*)d_in[8];
    const float* W_gat  = (const float*)d_in[9];
    const float* a_src  = (const float*)d_in[10];
    const float* a_dst  = (const float*)d_in[11];
    const float* b_gat  = (const float*)d_in[12];
    const float* W_cheb = (const float*)d_in[13];
    const float* b_cheb = (const float*)d_in[14];
    const float* W_gin  = (const float*)d_in[15];
    const float* b_gin  = (const float*)d_in[16];
    const ll*    eidx   = (const ll*)d_in[17];

    const ll* src = eidx;
    const ll* dst = eidx + NE;
    float* out = (float*)d_out;

    // workspace layout (all offsets 16B-aligned: N*4 = 200000 bytes)
    float* ws    = (float*)d_ws;
    float* deg   = ws;            // N
    float* dinv  = deg   + NN;    // N
    float* dinvc = dinv  + NN;    // N
    float* el    = dinvc + NN;    // N
    float* er    = el    + NN;    // N
    float* m     = er    + NN;    // N
    float* z     = m     + NN;    // N
    float* B1    = z     + NN;    // N*64  branch output
    float* B2    = B1 + (size_t)NN * DIM;  // N*64  nsum-derived / xw / T1
    float* B3    = B2 + (size_t)NN * DIM;  // N*64  h (gat), then T2 (cheb)

    const int n64 = NN * DIM;
    const int TB = 256;
    const int gN   = (NN  + TB - 1) / TB;
    const int g64  = (n64 + TB - 1) / TB;
    const int gE   = (NE  + TB - 1) / TB;
    const int E2   = NE + NN;
    const int gE2  = (E2  + TB - 1) / TB;
    const int gGemm = (NN + 127) / 128;  // 8 waves/block * 16 rows/wave

    // ---- degrees + neighbor sums ----
    k_fill<<<gN, TB, 0, stream>>>(deg, 0.0f, NN);
    k_fill<<<g64, TB, 0, stream>>>(B2, 0.0f, n64);  // B2 <- nsum
    k_deg_nsum<<<gE, TB, 0, stream>>>(src, dst, x, deg, B2, NE);
    k_dinv<<<gN, TB, 0, stream>>>(deg, dinv, dinvc, NN);
    float* nsum = B2;

    // ---- branch 0: MLP ----
    k_gemm64<<<gGemm, TB, 0, stream>>>(x, W_mlp, b_mlp, B1, NN, 0);
    k_combine<<<g64, TB, 0, stream>>>(out, B1, wts, 0, 1, n64);

    // ---- branch 1: SAGE (mean) ----  nmean -> B3 (keeps nsum in B2 alive)
    k_nmean<<<g64, TB, 0, stream>>>(nsum, deg, B3, n64);
    k_gemm64<<<gGemm, TB, 0, stream>>>(B3, Wl_sg, b_sg, B1, NN, 0);
    k_gemm64<<<gGemm, TB, 0, stream>>>(x, Wr_sg, nullptr, B1, NN, 1);
    k_combine<<<g64, TB, 0, stream>>>(out, B1, wts, 1, 0, n64);

    // ---- branch 5: GIN (uses nsum, then frees B2) ----
    k_addxn<<<g64, TB, 0, stream>>>(x, nsum, B3, n64);
    k_gemm64<<<gGemm, TB, 0, stream>>>(B3, W_gin, b_gin, B1, NN, 0);
    k_combine<<<g64, TB, 0, stream>>>(out, B1, wts, 5, 0, n64);

    // ---- branch 2: GCN ----  xw -> B2
    k_gemm64<<<gGemm, TB, 0, stream>>>(x, W_gcn, nullptr, B2, NN, 0);
    k_fill<<<g64, TB, 0, stream>>>(B1, 0.0f, n64);
    k_scatter_norm<<<gE, TB, 0, stream>>>(src, dst, B2, dinv, B1, NE);
    k_gcn_self<<<g64, TB, 0, stream>>>(B1, B2, dinv, b_gcn, n64);
    k_combine<<<g64, TB, 0, stream>>>(out, B1, wts, 2, 0, n64);

    // ---- branch 3: GAT ----  h -> B3
    k_gemm64<<<gGemm, TB, 0, stream>>>(x, W_gat, nullptr, B3, NN, 0);
    k_el_er<<<gN, TB, 0, stream>>>(B3, a_src, a_dst, el, er, NN);
    k_fill<<<gN, TB, 0, stream>>>(m, -1e30f, NN);
    k_fill<<<gN, TB, 0, stream>>>(z, 0.0f, NN);
    k_gat_max<<<gE2, TB, 0, stream>>>(src, dst, el, er, m, NE, E2);
    k_gat_sum<<<gE2, TB, 0, stream>>>(src, dst, el, er, m, z, NE, E2);
    k_fill<<<g64, TB, 0, stream>>>(B1, 0.0f, n64);
    k_gat_acc<<<gE2, TB, 0, stream>>>(src, dst, el, er, m, z, B3, B1, NE, E2);
    k_addbias<<<g64, TB, 0, stream>>>(B1, b_gat, n64);
    k_combine<<<g64, TB, 0, stream>>>(out, B1, wts, 3, 0, n64);

    // ---- branch 4: Cheb K=3 ----  T1 -> B2, T2 -> B3
    k_fill<<<g64, TB, 0, stream>>>(B2, 0.0f, n64);
    k_scatter_norm<<<gE, TB, 0, stream>>>(src, dst, x, dinvc, B2, NE);   // P0
    k_negate<<<g64, TB, 0, stream>>>(B2, n64);                           // T1 = -P0
    k_fill<<<g64, TB, 0, stream>>>(B3, 0.0f, n64);
    k_scatter_norm<<<gE, TB, 0, stream>>>(src, dst, B2, dinvc, B3, NE);  // P1 = prop(T1)
    k_t2<<<g64, TB, 0, stream>>>(B3, x, n64);                            // T2 = -2*P1 - x
    k_gemm64<<<gGemm, TB, 0, stream>>>(x,  W_cheb,            b_cheb,  B1, NN, 0);
    k_gemm64<<<gGemm, TB, 0, stream>>>(B2, W_cheb + 64 * 64,  nullptr, B1, NN, 1);
    k_gemm64<<<gGemm, TB, 0, stream>>>(B3, W_cheb + 2*64*64,  nullptr, B1, NN, 1);
    k_combine<<<g64, TB, 0, stream>>>(out, B1, wts, 4, 0, n64);
}